// PSAvgPooling_17188459118947
// MI455X (gfx1250) — compile-verified
//
#include <hip/hip_runtime.h>

// Position-sensitive ROI average pooling (R-FCN style) for MI455X / gfx1250.
//
// inputs : [8, 64, 64, 1152] f32   (viewed as [H, W, 9 bins, 128 ch])
// boxes  : [8, 512, 4]       f32   normalized (y1, x1, y2, x2)
// weights: [9]               f32
// out    : [4096, 128]       f32
//
// Design: gather-bound op (680 MB of 512B-contiguous gathers, input fits in
// 192MB L2). One block (96 threads = 3 wave32) per box; wave w does bin-row w
// with float4 (B128) per-lane gathers covering the full 128-ch slice per
// request. Per-bin sums staged in LDS, then the 9-bin weighted reduction is
// done with chained V_WMMA_F32_16X16X4_F32 (bins padded to 12 = 3 K-chunks),
// A = feats[16ch x 4bin] in the ISA A-layout, B = weights broadcast.

namespace {

constexpr int kH = 64, kW = 64, kC = 1152;
constexpr int kTB = 9;        // 3x3 bins
constexpr int kCs = 128;      // channels per bin
constexpr int kNB = 3, kCrop = 3;
constexpr int kBoxPerImg = 512;

typedef __attribute__((ext_vector_type(2))) float v2f;
typedef __attribute__((ext_vector_type(8))) float v8f;

__device__ __forceinline__ int iclamp(int v, int lo, int hi) {
  return v < lo ? lo : (v > hi ? hi : v);
}

__global__ __launch_bounds__(96) void ps_pool_wmma_kernel(
    const float* __restrict__ inputs, const float* __restrict__ boxes,
    const float* __restrict__ weights, float* __restrict__ out) {
  __shared__ float feats[12][kCs];  // bins 9..11 zero padding for WMMA K-chunks
  __shared__ float outb[kCs];

  const int lane = threadIdx.x & 31;
  const int wv   = threadIdx.x >> 5;   // 0..2 : bin row handled by this wave
  const int box  = blockIdx.x;         // one box per block
  const int c0   = lane * 4;           // this lane's 4 channels (float4)

  const float* __restrict__ im =
      inputs + (size_t)(box / kBoxPerImg) * (size_t)(kH * kW * kC);

  if (wv == 0) {
    const float4 z = make_float4(0.f, 0.f, 0.f, 0.f);
    *(float4*)&feats[9][c0]  = z;
    *(float4*)&feats[10][c0] = z;
    *(float4*)&feats[11][c0] = z;
  }

  const float4 b4 = ((const float4*)boxes)[box];  // y1, x1, y2, x2
  const float y1 = b4.x, x1 = b4.y, y2 = b4.z, x2 = b4.w;
  const float sy = (y2 - y1) * (1.f / 3.f);
  const float sx = (x2 - x1) * (1.f / 3.f);

  // Precompute the 9 x-sample quantities (bin_x * 3 + px); fully unrolled ->
  // lives in registers.
  int   xo0[9], xo1[9];
  float wxs[9], vxs[9];
#pragma unroll
  for (int bx = 0; bx < kNB; ++bx) {
#pragma unroll
    for (int px = 0; px < kCrop; ++px) {
      const int i = bx * kCrop + px;
      const float xc = (x1 + ((float)bx + 0.5f * (float)px) * sx) * (float)(kW - 1);
      const float xf = floorf(xc);
      const int  xi  = (int)xf;
      xo0[i] = iclamp(xi, 0, kW - 1) * (kTB * kCs);
      xo1[i] = iclamp(xi + 1, 0, kW - 1) * (kTB * kCs);
      wxs[i] = xc - xf;
      vxs[i] = (xc >= 0.f && xc <= (float)(kW - 1)) ? 1.f : 0.f;
    }
  }

  // Main gather loop: this wave's bin row (3 bins x 9 points x 4 corners).
  const int by = wv;
  float4 acc[3];
#pragma unroll
  for (int j = 0; j < 3; ++j) acc[j] = make_float4(0.f, 0.f, 0.f, 0.f);

#pragma unroll
  for (int py = 0; py < kCrop; ++py) {
    const float yc = (y1 + ((float)by + 0.5f * (float)py) * sy) * (float)(kH - 1);
    const float yf = floorf(yc);
    const int  yi  = (int)yf;
    const int  yo0 = iclamp(yi, 0, kH - 1) * (kW * kTB * kCs);
    const int  yo1 = iclamp(yi + 1, 0, kH - 1) * (kW * kTB * kCs);
    const float wy = yc - yf;
    const float vy = (yc >= 0.f && yc <= (float)(kH - 1)) ? 1.f : 0.f;
#pragma unroll
    for (int bx = 0; bx < kNB; ++bx) {
      const int boff = (by * kNB + bx) * kCs + c0;
#pragma unroll
      for (int px = 0; px < kCrop; ++px) {
        const int ix = bx * kCrop + px;
        // 4 independent 512B-per-wave gathers (good MLP, L2-resident)
        const float4 v00 = *(const float4*)(im + yo0 + xo0[ix] + boff);
        const float4 v01 = *(const float4*)(im + yo0 + xo1[ix] + boff);
        const float4 v10 = *(const float4*)(im + yo1 + xo0[ix] + boff);
        const float4 v11 = *(const float4*)(im + yo1 + xo1[ix] + boff);
        const float wx  = wxs[ix];
        const float msk = vy * vxs[ix];
        const float w00 = (1.f - wy) * (1.f - wx) * msk;
        const float w01 = (1.f - wy) * wx * msk;
        const float w10 = wy * (1.f - wx) * msk;
        const float w11 = wy * wx * msk;
        acc[bx].x += w00 * v00.x + w01 * v01.x + w10 * v10.x + w11 * v11.x;
        acc[bx].y += w00 * v00.y + w01 * v01.y + w10 * v10.y + w11 * v11.y;
        acc[bx].z += w00 * v00.z + w01 * v01.z + w10 * v10.z + w11 * v11.z;
        acc[bx].w += w00 * v00.w + w01 * v01.w + w10 * v10.w + w11 * v11.w;
      }
    }
  }
#pragma unroll
  for (int bx = 0; bx < kNB; ++bx)
    *(float4*)&feats[by * kNB + bx][c0] = acc[bx];

  __syncthreads();

  // ---- WMMA epilogue: out[c] = (1/81) * sum_bin w[bin] * feats[bin][c] ----
  // D(16x16) = A(16ch x 4bin) * B(4bin x 16) + C, chained over 3 K-chunks.
  // A layout (32-bit 16x4): VGPR0 = K0 (lanes 0-15) / K2 (lanes 16-31),
  //                         VGPR1 = K1 / K3.  B mirrors (rows K across halves).
  float wp[12];
#pragma unroll
  for (int j = 0; j < 12; ++j) wp[j] = (j < 9) ? weights[j] : 0.f;

  const int  half  = lane >> 4;  // which lane-half (selects K slots 0/1 vs 2/3)
  const int  m     = lane & 15;  // A-matrix row (channel within group)
  const float scale = 1.f / 81.f;

  for (int grp = wv; grp < 8; grp += 3) {  // 8 channel groups split over waves
    v8f d = {0.f, 0.f, 0.f, 0.f, 0.f, 0.f, 0.f, 0.f};
#pragma unroll
    for (int t = 0; t < 3; ++t) {
      v2f a, b;
      a[0] = feats[t * 4 + half * 2 + 0][grp * 16 + m];
      a[1] = feats[t * 4 + half * 2 + 1][grp * 16 + m];
      b[0] = half ? wp[t * 4 + 2] : wp[t * 4 + 0];
      b[1] = half ? wp[t * 4 + 3] : wp[t * 4 + 1];
      d = __builtin_amdgcn_wmma_f32_16x16x4_f32(
          /*neg_a=*/false, a, /*neg_b=*/false, b,
          /*c_mod=*/(short)0, d, /*reuse_a=*/false, /*reuse_b=*/false);
    }
    // D VGPR r holds (row r, lanes 0-15) / (row 8+r, lanes 16-31); all columns
    // identical since B columns are identical. Lane (half*16 + sel), sel<8,
    // provides row half*8+sel from d[sel].
    const int sel = lane & 7;
    float tmp = d[0];
    if (sel == 1) tmp = d[1];
    if (sel == 2) tmp = d[2];
    if (sel == 3) tmp = d[3];
    if (sel == 4) tmp = d[4];
    if (sel == 5) tmp = d[5];
    if (sel == 6) tmp = d[6];
    if (sel == 7) tmp = d[7];
    if (m < 8) outb[grp * 16 + half * 8 + sel] = tmp * scale;
  }

  __syncthreads();

  if (threadIdx.x < 32)
    ((float4*)(out + (size_t)box * kCs))[threadIdx.x] =
        ((const float4*)outb)[threadIdx.x];
}

}  // namespace

extern "C" void kernel_launch(void* const* d_in, const int* in_sizes, int n_in,
                              void* d_out, int out_size, void* d_ws, size_t ws_size,
                              hipStream_t stream) {
  const float* inputs  = (const float*)d_in[0];
  const float* boxes   = (const float*)d_in[1];
  const float* weights = (const float*)d_in[2];
  float* out = (float*)d_out;

  const int n_boxes = in_sizes[1] / 4;  // 8 * 512 = 4096
  dim3 grid(n_boxes), block(96);        // 3 wave32 per box (one per bin row)
  hipLaunchKernelGGL(ps_pool_wmma_kernel, grid, block, 0, stream,
                     inputs, boxes, weights, out);
}